// Attn_69114613729280
// MI455X (gfx1250) — compile-verified
//
#include <hip/hip_runtime.h>
#include <math.h>

#define HID 2048
#define SEQ 16384

typedef __attribute__((ext_vector_type(2))) float v2f;
typedef __attribute__((ext_vector_type(4))) float v4f;
typedef __attribute__((ext_vector_type(8))) float v8f;

// ---------------------------------------------------------------------------
// Phase 1a: partial[dt][h] = sum over d-tile dt of W[d][h] * hidden[d]
// Grid: 64 blocks (2 h-tiles x 32 d-tiles), 256 threads, float4 per thread.
// Fully coalesced b128 streaming of W (16 MB).
// ---------------------------------------------------------------------------
__global__ void k_wT_partial(const float* __restrict__ W,
                             const float* __restrict__ hidden,
                             float* __restrict__ partial) {
  const int ht = blockIdx.x & 1;    // 2 h-tiles of 1024 floats
  const int dt = blockIdx.x >> 1;   // 32 d-tiles of 64 rows
  const int h4 = ht * 1024 + threadIdx.x * 4;
  const int d0 = dt * 64;
  v4f acc = {};
#pragma unroll 4
  for (int d = 0; d < 64; ++d) {
    v4f w = *(const v4f*)(W + (size_t)(d0 + d) * HID + h4);
    const float hv = hidden[d0 + d];
    acc.x = fmaf(w.x, hv, acc.x);
    acc.y = fmaf(w.y, hv, acc.y);
    acc.z = fmaf(w.z, hv, acc.z);
    acc.w = fmaf(w.w, hv, acc.w);
  }
  *(v4f*)(partial + dt * HID + h4) = acc;
}

// ---------------------------------------------------------------------------
// Phase 1b: v[h] = sum_dt partial[dt][h]   (deterministic, no atomics)
// Grid: 2 blocks, 256 threads, float4 per thread.
// ---------------------------------------------------------------------------
__global__ void k_wT_reduce(const float* __restrict__ partial,
                            float* __restrict__ v) {
  const int h4 = blockIdx.x * 1024 + threadIdx.x * 4;
  v4f s = {};
#pragma unroll
  for (int dt = 0; dt < 32; ++dt) {
    v4f p = *(const v4f*)(partial + dt * HID + h4);
    s.x += p.x; s.y += p.y; s.z += p.z; s.w += p.w;
  }
  *(v4f*)(v + h4) = s;
}

// ---------------------------------------------------------------------------
// Phase 2: score[s] = enc[s,:] . v  via V_WMMA_F32_16X16X4_F32.
// Each wave32 computes 16 rows. Per 16-column chunk, lane (m, half) loads
// two float4 of enc row m at cols k0+8*half (global_load_b128) and the
// matching two float4 of v from LDS. WMMA j consumes the aligned v2f
// sub-pairs (2j,2j+1) of each float4; A and B use the identical
// (half,VGPR)->column placement, so D[m][n] = enc_row_m . v_chunk for all n
// (columns are merely visited in a permuted order, which is sum-invariant).
// Grid: 256 blocks x 128 threads (4 waves, 64 rows per block).
// ---------------------------------------------------------------------------
__global__ void k_scores_wmma(const float* __restrict__ enc,
                              const float* __restrict__ v,
                              float* __restrict__ scores) {
  __shared__ float vs[HID];
  const int tid = threadIdx.x;
  {
    v4f* vs4 = (v4f*)vs;
    const v4f* v4 = (const v4f*)v;
    for (int i = tid; i < HID / 4; i += 128) vs4[i] = v4[i];
  }
  __syncthreads();

  const int wave = tid >> 5;
  const int lane = tid & 31;
  const int m    = lane & 15;
  const int half = lane >> 4;
  const int row_base = blockIdx.x * 64 + wave * 16;

  const float* rowp = enc + (size_t)(row_base + m) * HID + 8 * half;
  const float* vp   = vs + 8 * half;

  v8f acc = {};
  for (int k0 = 0; k0 < HID; k0 += 16) {
    v4f a0 = *(const v4f*)(rowp + k0);       // global_load_b128 (streaming enc)
    v4f a1 = *(const v4f*)(rowp + k0 + 4);
    v4f b0 = *(const v4f*)(vp + k0);         // ds_load_b128 (broadcast)
    v4f b1 = *(const v4f*)(vp + k0 + 4);

    v2f a0lo = __builtin_shufflevector(a0, a0, 0, 1);
    v2f a0hi = __builtin_shufflevector(a0, a0, 2, 3);
    v2f a1lo = __builtin_shufflevector(a1, a1, 0, 1);
    v2f a1hi = __builtin_shufflevector(a1, a1, 2, 3);
    v2f b0lo = __builtin_shufflevector(b0, b0, 0, 1);
    v2f b0hi = __builtin_shufflevector(b0, b0, 2, 3);
    v2f b1lo = __builtin_shufflevector(b1, b1, 0, 1);
    v2f b1hi = __builtin_shufflevector(b1, b1, 2, 3);

    acc = __builtin_amdgcn_wmma_f32_16x16x4_f32(false, a0lo, false, b0lo,
                                                (short)0, acc, false, false);
    acc = __builtin_amdgcn_wmma_f32_16x16x4_f32(false, a0hi, false, b0hi,
                                                (short)0, acc, false, false);
    acc = __builtin_amdgcn_wmma_f32_16x16x4_f32(false, a1lo, false, b1lo,
                                                (short)0, acc, false, false);
    acc = __builtin_amdgcn_wmma_f32_16x16x4_f32(false, a1hi, false, b1hi,
                                                (short)0, acc, false, false);
  }

  // C/D layout: VGPR i holds M=i (lanes 0-15) and M=i+8 (lanes 16-31);
  // every column identical -> lane 0 has rows 0..7, lane 16 has rows 8..15.
  v4f lo = __builtin_shufflevector(acc, acc, 0, 1, 2, 3);
  v4f hi = __builtin_shufflevector(acc, acc, 4, 5, 6, 7);
  if (lane == 0) {
    *(v4f*)(scores + row_base)     = lo;
    *(v4f*)(scores + row_base + 4) = hi;
  } else if (lane == 16) {
    *(v4f*)(scores + row_base + 8)  = lo;
    *(v4f*)(scores + row_base + 12) = hi;
  }
}

// ---------------------------------------------------------------------------
// Phase 3: softmax over 16384 scores. Single block, 1024 threads (32 waves).
// The reference's "+ b" bias adds the constant b.hidden to every score and
// cancels in softmax, so it is intentionally omitted.
// ---------------------------------------------------------------------------
__global__ void k_softmax(const float* __restrict__ scores,
                          float* __restrict__ out) {
  __shared__ float red[1024];
  const int tid = threadIdx.x;

  float m = -INFINITY;
  for (int i = tid; i < SEQ; i += 1024) m = fmaxf(m, scores[i]);
  red[tid] = m;
  __syncthreads();
  for (int s = 512; s > 0; s >>= 1) {
    if (tid < s) red[tid] = fmaxf(red[tid], red[tid + s]);
    __syncthreads();
  }
  const float gmax = red[0];
  __syncthreads();

  float sum = 0.f;
  for (int i = tid; i < SEQ; i += 1024) sum += __expf(scores[i] - gmax);
  red[tid] = sum;
  __syncthreads();
  for (int s = 512; s > 0; s >>= 1) {
    if (tid < s) red[tid] += red[tid + s];
    __syncthreads();
  }
  const float inv = 1.0f / red[0];

  for (int i = tid; i < SEQ; i += 1024) out[i] = __expf(scores[i] - gmax) * inv;
}

// ---------------------------------------------------------------------------
extern "C" void kernel_launch(void* const* d_in, const int* in_sizes, int n_in,
                              void* d_out, int out_size, void* d_ws, size_t ws_size,
                              hipStream_t stream) {
  const float* hidden = (const float*)d_in[0];   // [1, 2048]
  const float* enc    = (const float*)d_in[1];   // [16384, 2048]
  const float* W      = (const float*)d_in[2];   // [2048, 2048]
  // d_in[3] = b : cancels in softmax, unused.
  float* out = (float*)d_out;                    // [16384]

  float* ws      = (float*)d_ws;
  float* partial = ws;                 // 32 * 2048 floats (256 KB)
  float* v       = ws + 32 * HID;      // 2048 floats (8 KB)
  float* scores  = v + HID;            // 16384 floats (64 KB)

  k_wT_partial <<<64, 256, 0, stream>>>(W, hidden, partial);
  k_wT_reduce  <<<2,  256, 0, stream>>>(partial, v);
  k_scores_wmma<<<256, 128, 0, stream>>>(enc, v, scores);
  k_softmax    <<<1, 1024, 0, stream>>>(scores, out);
}